// Users_sets_Encoder_51092930953378
// MI455X (gfx1250) — compile-verified
//
#include <hip/hip_runtime.h>
#include <hip/hip_bf16.h>

// Problem constants (match reference)
#define GG   8
#define LL   50
#define SS   32
#define DD   64
#define ATT  16
#define ROWS 16           // samples per block (== WMMA M tile)
#define COMB_STRIDE 132   // 128 + 4 pad -> conflict-free A-frag reads
#define MEM_STRIDE  66    // 64 + 2 pad  -> conflict-free member reads

typedef __attribute__((ext_vector_type(2))) float v2f;
typedef __attribute__((ext_vector_type(8))) float v8f;

__global__ __launch_bounds__(512) void users_sets_encoder_kernel(
    const int* __restrict__ nodes,            // [B,8]
    const int* __restrict__ hist_u,           // [B,50]
    const int* __restrict__ hist_r,           // [B,50]
    const unsigned char* __restrict__ hist_m, // [B,50] bool
    const int* __restrict__ soc_adj,          // [B,32]
    const unsigned char* __restrict__ soc_m,  // [B,32] bool
    const float* __restrict__ features,       // [NUM_USERS,64]
    const float* __restrict__ r_embed,        // [5,64]
    const float* __restrict__ wg1,            // [64,16]
    const float* __restrict__ bg1,            // [16]
    const float* __restrict__ wg2,            // [16,1]
    const float* __restrict__ bg2,            // [1]
    const float* __restrict__ w1,             // [128,64]
    const float* __restrict__ b1,             // [64]
    float* __restrict__ out)                  // [B,64]
{
    __shared__ float mem_sh[ROWS][GG][MEM_STRIDE];   // staged member embeddings
    __shared__ float comb_sh[ROWS][COMB_STRIDE];     // concat(self, neigh) per row
    __shared__ float logit_sh[ROWS][GG];
    __shared__ float wg1_sh[DD * ATT];               // 4 KB, block-resident

    const int tid  = threadIdx.x;
    const int wave = tid >> 5;      // 0..15 : one sample per wave
    const int lane = tid & 31;
    const int b0   = blockIdx.x * ROWS;
    const int b    = b0 + wave;

    // cooperative copy of wg1 into LDS (1024 floats / 512 threads)
    wg1_sh[tid]       = wg1[tid];
    wg1_sh[tid + 512] = wg1[tid + 512];

    // ---------------- Phase 1: coalesced gathers (lane owns dims 2l, 2l+1) ----
    const int d0 = 2 * lane;
    float2 m[GG];
    #pragma unroll
    for (int g = 0; g < GG; ++g) {
        const int idx = nodes[b * GG + g];
        const float2 v = *(const float2*)(features + (size_t)idx * DD + d0);
        m[g] = v;
        mem_sh[wave][g][d0]     = v.x;
        mem_sh[wave][g][d0 + 1] = v.y;
    }

    // Branchless masked means: all indices are valid, so issue every gather
    // unconditionally and weight by the mask -> straight-line clause'd loads,
    // L2 latency hidden by many loads in flight instead of branch-serialized.
    float hx = 0.f, hy = 0.f, hcf = 0.f;
    #pragma unroll 5
    for (int j = 0; j < LL; ++j) {
        const float mv = (float)hist_m[b * LL + j];
        const int   iu = hist_u[b * LL + j];
        const int   ir = hist_r[b * LL + j];
        const float2 f = *(const float2*)(features + (size_t)iu * DD + d0);
        const float2 r = *(const float2*)(r_embed + ir * DD + d0);
        hx  += mv * (f.x + r.x);
        hy  += mv * (f.y + r.y);
        hcf += mv;
    }
    const float hden = 1.f / fmaxf(hcf, 1.f);

    float sx = 0.f, sy = 0.f, scf = 0.f;
    #pragma unroll 4
    for (int j = 0; j < SS; ++j) {
        const float mv = (float)soc_m[b * SS + j];
        const int   is = soc_adj[b * SS + j];
        const float2 f = *(const float2*)(features + (size_t)is * DD + d0);
        sx  += mv * f.x;
        sy  += mv * f.y;
        scf += mv;
    }
    const float sden = 1.f / fmaxf(scf, 1.f);
    const float nx = 0.5f * (hx * hden + sx * sden);
    const float ny = 0.5f * (hy * hden + sy * sden);

    __syncthreads();   // wg1_sh ready (members are same-wave, already ordered)

    // ---------------- Phase 2: attention. lane -> (g = lane>>2, a = (lane&3)*4+i)
    const int ag = lane >> 2;
    const int aq = lane & 3;
    float part = 0.f;
    #pragma unroll
    for (int i = 0; i < 4; ++i) {
        const int a = aq * 4 + i;
        float acc = bg1[a];
        #pragma unroll 8
        for (int d = 0; d < DD; ++d)
            acc += mem_sh[wave][ag][d] * wg1_sh[d * ATT + a];
        part += tanhf(acc) * wg2[a];
    }
    // reduce the 4 lanes of this g-group
    part += __shfl_xor(part, 1);
    part += __shfl_xor(part, 2);
    part += bg2[0];
    if (aq == 0) logit_sh[wave][ag] = part;  // same-wave LDS: ordered, no barrier

    float lg[GG];
    float lmax = -3.0e38f;
    #pragma unroll
    for (int g = 0; g < GG; ++g) {
        lg[g] = logit_sh[wave][g];
        lmax = fmaxf(lmax, lg[g]);
    }
    float lsum = 0.f;
    #pragma unroll
    for (int g = 0; g < GG; ++g) {
        lg[g] = __expf(lg[g] - lmax);
        lsum += lg[g];
    }
    const float inv = 1.f / lsum;

    float selfx = 0.f, selfy = 0.f;
    #pragma unroll
    for (int g = 0; g < GG; ++g) {
        const float a = lg[g] * inv;
        selfx += a * m[g].x;
        selfy += a * m[g].y;
    }

    comb_sh[wave][d0]          = selfx;
    comb_sh[wave][d0 + 1]      = selfy;
    comb_sh[wave][DD + d0]     = nx;
    comb_sh[wave][DD + d0 + 1] = ny;

    __syncthreads();   // full 16x128 combined tile visible to GEMM waves

    // ---------------- Phase 3: WMMA fp32 GEMM  [16x128] @ [128x64] -------------
    if (wave < 4) {
        const int n0  = wave * 16;      // output column tile
        const int row = lane & 15;      // M index for A, N index for B/D
        const int hi  = lane >> 4;      // K-half selector per ISA layout
        v8f dacc = {};
        #pragma unroll 8
        for (int k = 0; k < 128; k += 4) {
            const int ka = k + 2 * hi;
            v2f afrag, bfrag;
            afrag.x = comb_sh[row][ka];
            afrag.y = comb_sh[row][ka + 1];
            bfrag.x = w1[ka * DD + n0 + row];
            bfrag.y = w1[(ka + 1) * DD + n0 + row];
            dacc = __builtin_amdgcn_wmma_f32_16x16x4_f32(
                false, afrag, false, bfrag, (short)0, dacc, false, false);
        }
        #pragma unroll
        for (int v = 0; v < 8; ++v) {
            const int mrow = v + 8 * hi;         // C/D layout: M = v (+8 for hi lanes)
            const int col  = n0 + row;           // N = lane & 15
            const float val = dacc[v] + b1[col];
            out[(size_t)(b0 + mrow) * DD + col] = fmaxf(val, 0.f);
        }
    }
}

extern "C" void kernel_launch(void* const* d_in, const int* in_sizes, int n_in,
                              void* d_out, int out_size, void* d_ws, size_t ws_size,
                              hipStream_t stream) {
    (void)n_in; (void)d_ws; (void)ws_size; (void)out_size;
    const int* nodes           = (const int*)d_in[0];
    const int* hist_u          = (const int*)d_in[1];
    const int* hist_r          = (const int*)d_in[2];
    const unsigned char* hmask = (const unsigned char*)d_in[3];
    const int* soc_adj         = (const int*)d_in[4];
    const unsigned char* smask = (const unsigned char*)d_in[5];
    const float* features      = (const float*)d_in[6];
    const float* r_embed       = (const float*)d_in[7];
    const float* wg1           = (const float*)d_in[8];
    const float* bg1           = (const float*)d_in[9];
    const float* wg2           = (const float*)d_in[10];
    const float* bg2           = (const float*)d_in[11];
    const float* w1            = (const float*)d_in[12];
    const float* b1            = (const float*)d_in[13];
    float* out                 = (float*)d_out;

    const int B = in_sizes[0] / GG;           // 16384
    dim3 grid(B / ROWS), block(512);
    users_sets_encoder_kernel<<<grid, block, 0, stream>>>(
        nodes, hist_u, hist_r, hmask, soc_adj, smask,
        features, r_embed, wg1, bg1, wg2, bg2, w1, b1, out);
}